// GCNLayerConfigurableMLP_73272142070247
// MI455X (gfx1250) — compile-verified
//
#include <hip/hip_runtime.h>

#define D 64

typedef __attribute__((ext_vector_type(2))) float v2f;
typedef __attribute__((ext_vector_type(8))) float v8f;

// ---------------------------------------------------------------------------
// One-time edge preprocessing
// ---------------------------------------------------------------------------
__global__ void zero_u32_kernel(unsigned* __restrict__ p, int n) {
    int i = blockIdx.x * blockDim.x + threadIdx.x;
    if (i < n) p[i] = 0u;
}

__global__ void prep_edges_kernel(const long long* __restrict__ ei, int E,
                                  int* __restrict__ rows, int* __restrict__ cols,
                                  unsigned* __restrict__ deg) {
    int e = blockIdx.x * blockDim.x + threadIdx.x;
    if (e >= E) return;
    int r = (int)ei[e];
    int c = (int)ei[(long long)E + e];
    rows[e] = r;
    cols[e] = c;
    atomicAdd(&deg[c], 1u);
}

__global__ void dinv_kernel(const unsigned* __restrict__ deg,
                            float* __restrict__ dinv, int n) {
    int i = blockIdx.x * blockDim.x + threadIdx.x;
    if (i < n) dinv[i] = rsqrtf((float)(deg[i] + 1u));  // +1 self loop; always > 0
}

__global__ void norm_kernel(const int* __restrict__ rows, const int* __restrict__ cols,
                            const float* __restrict__ dinv, float* __restrict__ normv,
                            int E) {
    int e = blockIdx.x * blockDim.x + threadIdx.x;
    if (e < E) normv[e] = dinv[rows[e]] * dinv[cols[e]];
}

// ---------------------------------------------------------------------------
// GEMM: H[M x 64] = X[M x 64] * W[64 x 64], fp32 via V_WMMA_F32_16X16X4_F32.
// One wave computes one 16x16 tile of H; K-loop of 16 steps of k=4.
// A frag (16x4 f32, 2 VGPR): lanes 0-15 -> M=lane, {K=k0,k0+1}; lanes 16-31 -> {K=k0+2,k0+3}
// B frag (4x16 f32, 2 VGPR): mirrored, N=lane&15.
// C/D (16x16 f32, 8 VGPR): vgpr v -> M = v + (lane<16?0:8), N = lane&15.
// ---------------------------------------------------------------------------
__global__ void gemm64_wmma_kernel(const float* __restrict__ X,
                                   const float* __restrict__ W,
                                   float* __restrict__ H, int mtiles) {
    int wave = (blockIdx.x * blockDim.x + threadIdx.x) >> 5;
    int lane = threadIdx.x & 31;
    int mt = wave >> 2;   // D/16 = 4 N-tiles
    int nt = wave & 3;
    if (mt >= mtiles) return;            // wave-uniform: EXEC stays all-1s for WMMA

    int half = lane >> 4;                // 0 or 1
    int l16  = lane & 15;
    int arow = mt * 16 + l16;            // A row for this lane
    int bcol = nt * 16 + l16;            // B/D column for this lane

    v8f acc = {};
    #pragma unroll
    for (int k0 = 0; k0 < D; k0 += 4) {
        int ka = k0 + half * 2;
        v2f a = *(const v2f*)(X + (size_t)arow * D + ka);   // K=ka, ka+1
        v2f b;
        b.x = W[(size_t)ka * D + bcol];                     // K=ka   row of W
        b.y = W[(size_t)(ka + 1) * D + bcol];               // K=ka+1 row of W
        acc = __builtin_amdgcn_wmma_f32_16x16x4_f32(
            /*neg_a=*/false, a, /*neg_b=*/false, b,
            /*c_mod=*/(short)0, acc, /*reuse_a=*/false, /*reuse_b=*/false);
    }

    int rbase = mt * 16 + half * 8;
    #pragma unroll
    for (int v = 0; v < 8; ++v)
        H[(size_t)(rbase + v) * D + bcol] = acc[v];
}

// out[i,:] = bias + dinv[i]^2 * h[i,:]   (self-loop term; also initializes out)
__global__ void init_out_kernel(const float* __restrict__ h,
                                const float* __restrict__ dinv,
                                const float* __restrict__ bias,
                                float* __restrict__ out, int n) {
    int t = blockIdx.x * blockDim.x + threadIdx.x;
    if (t >= n * (D / 4)) return;
    int i = t >> 4;            // node
    int f = (t & 15) * 4;      // feature quad
    float s = dinv[i];
    s *= s;
    float4 hv = *(const float4*)(h + (size_t)i * D + f);
    float4 bv = *(const float4*)(bias + f);
    float4 o;
    o.x = bv.x + s * hv.x;
    o.y = bv.y + s * hv.y;
    o.z = bv.z + s * hv.z;
    o.w = bv.w + s * hv.w;
    *(float4*)(out + (size_t)i * D + f) = o;
}

// per-edge: out[col,:] += norm * h[row,:]  (32 lanes/edge, float2 each, HW f32 atomics)
__global__ void scatter_edges_kernel(const float* __restrict__ h,
                                     const int* __restrict__ rows,
                                     const int* __restrict__ cols,
                                     const float* __restrict__ normv,
                                     float* __restrict__ out, int E) {
    int t = blockIdx.x * blockDim.x + threadIdx.x;
    int e = t >> 5;
    if (e >= E) return;
    int lane = t & 31;
    int r = rows[e];
    int c = cols[e];
    float nv = normv[e];
    int f = lane * 2;
    float2 hv = *(const float2*)(h + (size_t)r * D + f);
    unsafeAtomicAdd(out + (size_t)c * D + f,     nv * hv.x);
    unsafeAtomicAdd(out + (size_t)c * D + f + 1, nv * hv.y);
}

// ---------------------------------------------------------------------------
extern "C" void kernel_launch(void* const* d_in, const int* in_sizes, int n_in,
                              void* d_out, int out_size, void* d_ws, size_t ws_size,
                              hipStream_t stream) {
    const float*     x  = (const float*)d_in[0];
    const long long* ei = (const long long*)d_in[1];
    const float* Ws[3] = {(const float*)d_in[2], (const float*)d_in[4], (const float*)d_in[6]};
    const float* Bs[3] = {(const float*)d_in[3], (const float*)d_in[5], (const float*)d_in[7]};

    const int N = in_sizes[0] / D;
    const int E = in_sizes[1] / 2;

    // workspace carve-up (256B aligned regions)
    char* ws = (char*)d_ws;
    size_t off = 0;
    auto carve = [&](size_t bytes) -> void* {
        void* p = ws + off;
        off += (bytes + 255) & ~(size_t)255;
        return p;
    };
    unsigned* deg    = (unsigned*)carve((size_t)N * 4);
    float*    dinv   = (float*)   carve((size_t)N * 4);
    int*      rows32 = (int*)     carve((size_t)E * 4);
    int*      cols32 = (int*)     carve((size_t)E * 4);
    float*    normv  = (float*)   carve((size_t)E * 4);
    float*    hbuf   = (float*)   carve((size_t)N * D * 4);
    float*    agg1   = (float*)   carve((size_t)N * D * 4);
    float*    agg2   = (float*)   carve((size_t)N * D * 4);

    const int B = 256;

    // --- one-time edge preprocessing (degrees identical across all 3 layers)
    zero_u32_kernel<<<(N + B - 1) / B, B, 0, stream>>>(deg, N);
    prep_edges_kernel<<<(E + B - 1) / B, B, 0, stream>>>(ei, E, rows32, cols32, deg);
    dinv_kernel<<<(N + B - 1) / B, B, 0, stream>>>(deg, dinv, N);
    norm_kernel<<<(E + B - 1) / B, B, 0, stream>>>(rows32, cols32, dinv, normv, E);

    // --- 3 GCN layers
    const int mtiles      = (N + 15) / 16;            // 50000/16 = 3125 exact
    const int gemm_waves  = mtiles * (D / 16);        // 4 N-tiles
    const int gemm_blocks = (gemm_waves + 7) / 8;     // 8 waves / 256-thread block
    const int init_thr    = N * (D / 4);
    const int scat_thr    = E * 32;

    float* layer_out[3] = {agg1, agg2, (float*)d_out};
    const float* X = x;
    for (int l = 0; l < 3; ++l) {
        gemm64_wmma_kernel<<<gemm_blocks, B, 0, stream>>>(X, Ws[l], hbuf, mtiles);
        init_out_kernel<<<(init_thr + B - 1) / B, B, 0, stream>>>(hbuf, dinv, Bs[l],
                                                                  layer_out[l], N);
        scatter_edges_kernel<<<((size_t)scat_thr + B - 1) / B, B, 0, stream>>>(
            hbuf, rows32, cols32, normv, layer_out[l], E);
        X = layer_out[l];
    }
}